// PerforantHebb_AHA_87144886436114
// MI455X (gfx1250) — compile-verified
//
#include <hip/hip_runtime.h>
#include <hip/hip_bf16.h>
#include <stdint.h>

// Problem constants (match reference)
#define BATCH 1024
#define EC    4096
#define CA3   2048
#define LR    0.01f

typedef float v2f __attribute__((ext_vector_type(2)));
typedef float v8f __attribute__((ext_vector_type(8)));

static __device__ __forceinline__ v8f wmma_f32_k4(v2f a, v2f b, v8f c) {
    // D = A(16x4,f32) * B(4x16,f32) + C(16x16,f32)
    return __builtin_amdgcn_wmma_f32_16x16x4_f32(
        false, a, false, b, (short)0, c, false, false);
}

// One K-step (4 deep) of the row-major x row-major^T GEMM.
// ak/bk: uniform scalar bases already advanced to k0; avoff/bvoff: per-lane
// 32-bit element offsets; tile offsets i*16*EC become instruction immediates.
static __device__ __forceinline__ void gemm_step_rowmajor(
    const float* ak, const float* bk, uint32_t avoff, uint32_t bvoff,
    v8f (&acc)[4][4])
{
    v2f a[4], b[4];
#pragma unroll
    for (int i = 0; i < 4; ++i)
        a[i] = *(const v2f*)(ak + avoff + (size_t)i * 16 * EC);
#pragma unroll
    for (int i = 0; i < 4; ++i)
        b[i] = *(const v2f*)(bk + bvoff + (size_t)i * 16 * EC);
#pragma unroll
    for (int mi = 0; mi < 4; ++mi)
#pragma unroll
        for (int ni = 0; ni < 4; ++ni)
            acc[mi][ni] = wmma_f32_k4(a[mi], b[ni], acc[mi][ni]);
}

// ---------------------------------------------------------------------------
// Kernel 1: pre_pc_cue = ec @ W^T + dg        (M=BATCH, N=CA3, K=EC)
// Block: 256 thr = 8 waves as 2(M) x 4(N). Wave tile 64x64 = 4x4 WMMA tiles.
// Block tile 128x256. Grid (CA3/256, BATCH/128).
// ---------------------------------------------------------------------------
__global__ __launch_bounds__(256) void k_gemm_pre(
    const float* __restrict__ ec, const float* __restrict__ dg,
    const float* __restrict__ W, float* __restrict__ out)
{
    const int lane = threadIdx.x & 31;
    const int wave = threadIdx.x >> 5;
    const int wm = wave >> 2;            // 0..1
    const int wn = wave & 3;             // 0..3
    const int lm = lane & 15;            // M / N index within tile
    const int lk = (lane >> 4) << 1;     // K pair offset: 0 or 2

    const int row0 = blockIdx.y * 128 + wm * 64;   // batch rows
    const int col0 = blockIdx.x * 256 + wn * 64;   // ca3 cols

    const uint32_t voff = (uint32_t)(lm * EC + lk);    // per-lane, loop-fixed
    const float* abase = ec + (size_t)row0 * EC;       // uniform scalar base
    const float* bbase = W  + (size_t)col0 * EC;       // uniform scalar base

    v8f acc[4][4];
#pragma unroll
    for (int i = 0; i < 4; ++i)
#pragma unroll
        for (int j = 0; j < 4; ++j) acc[i][j] = (v8f){};

    // Outer step = 32 floats (one 128B line): prefetch once per line.
    for (int kb = 0; kb < EC; kb += 32) {
        if (kb + 512 < EC) {
            __builtin_prefetch(abase + voff + kb + 512, 0, 1);
            __builtin_prefetch(bbase + voff + kb + 512, 0, 1);
        }
#pragma unroll
        for (int kk = 0; kk < 32; kk += 4)
            gemm_step_rowmajor(abase + kb + kk, bbase + kb + kk,
                               voff, voff, acc);
    }

    // C/D layout: VGPR r, lanes 0-15 -> M=r, lanes 16-31 -> M=r+8; N = lane&15
    const int rsel = (lane >> 4) * 8;
#pragma unroll
    for (int mi = 0; mi < 4; ++mi)
#pragma unroll
        for (int ni = 0; ni < 4; ++ni)
#pragma unroll
            for (int r = 0; r < 8; ++r) {
                const int row = row0 + mi * 16 + rsel + r;
                const int col = col0 + ni * 16 + lm;
                const size_t idx = (size_t)row * CA3 + col;
                out[idx] = acc[mi][ni][r] + dg[idx];
            }
}

// ---------------------------------------------------------------------------
// Kernel 2: colmean[t] = mean_b pc[b,t]
// ---------------------------------------------------------------------------
__global__ __launch_bounds__(256) void k_colmean(
    const float* __restrict__ pc, float* __restrict__ colmean)
{
    const int t = blockIdx.x * blockDim.x + threadIdx.x;
    if (t < CA3) {
        float s = 0.f;
        for (int b = 0; b < BATCH; ++b) s += pc[(size_t)b * CA3 + t];
        colmean[t] = s * (1.0f / BATCH);
    }
}

// ---------------------------------------------------------------------------
// Kernel 3: delta[t,i] = LR * ( (pc^T @ ec)[t,i]/B - colmean[t]*W[t,i] )
// GEMM M=CA3(t), N=EC(i), K=BATCH(b). A[m,k]=pc[k,t0+m], B[k,n]=ec[k,i0+n].
// K runs along rows here, so operands are two strided b32 loads each.
// Grid (EC/256, CA3/128).
// ---------------------------------------------------------------------------
__global__ __launch_bounds__(256) void k_delta(
    const float* __restrict__ pc, const float* __restrict__ ec,
    const float* __restrict__ W, const float* __restrict__ colmean,
    float* __restrict__ delta)
{
    const int lane = threadIdx.x & 31;
    const int wave = threadIdx.x >> 5;
    const int wm = wave >> 2;
    const int wn = wave & 3;
    const int lm = lane & 15;
    const int lk = (lane >> 4) << 1;

    const int row0 = blockIdx.y * 128 + wm * 64;   // t
    const int col0 = blockIdx.x * 256 + wn * 64;   // i

    const uint32_t avoff = (uint32_t)(lk * CA3 + lm);  // per-lane, loop-fixed
    const uint32_t bvoff = (uint32_t)(lk * EC  + lm);
    const float* abase = pc + row0;                    // uniform scalar base
    const float* bbase = ec + col0;

    v8f acc[4][4];
#pragma unroll
    for (int i = 0; i < 4; ++i)
#pragma unroll
        for (int j = 0; j < 4; ++j) acc[i][j] = (v8f){};

#pragma unroll 2
    for (int k0 = 0; k0 < BATCH; k0 += 4) {
        const float* ak = abase + (size_t)k0 * CA3;    // scalar K-advance
        const float* bk = bbase + (size_t)k0 * EC;
        v2f a[4], b[4];
#pragma unroll
        for (int i = 0; i < 4; ++i) {
            const float* p = ak + avoff + i * 16;      // tile -> immediates
            a[i] = (v2f){ p[0], p[CA3] };
        }
#pragma unroll
        for (int i = 0; i < 4; ++i) {
            const float* p = bk + bvoff + i * 16;
            b[i] = (v2f){ p[0], p[EC] };
        }
#pragma unroll
        for (int mi = 0; mi < 4; ++mi)
#pragma unroll
            for (int ni = 0; ni < 4; ++ni)
                acc[mi][ni] = wmma_f32_k4(a[mi], b[ni], acc[mi][ni]);
    }

    const int rsel = (lane >> 4) * 8;
#pragma unroll
    for (int mi = 0; mi < 4; ++mi)
#pragma unroll
        for (int ni = 0; ni < 4; ++ni)
#pragma unroll
            for (int r = 0; r < 8; ++r) {
                const int t = row0 + mi * 16 + rsel + r;
                const int i = col0 + ni * 16 + lm;
                const size_t idx = (size_t)t * EC + i;
                delta[idx] = LR * (acc[mi][ni][r] * (1.0f / BATCH)
                                   - colmean[t] * W[idx]);
            }
}

// ---------------------------------------------------------------------------
// Kernel 4: diff = ec @ delta^T  (same shape as kernel 1); accumulate
// sum(diff^2) -> deterministic LDS tree reduce -> per-block partial.
// ---------------------------------------------------------------------------
__global__ __launch_bounds__(256) void k_loss(
    const float* __restrict__ ec, const float* __restrict__ delta,
    float* __restrict__ partials)
{
    const int lane = threadIdx.x & 31;
    const int wave = threadIdx.x >> 5;
    const int wm = wave >> 2;
    const int wn = wave & 3;
    const int lm = lane & 15;
    const int lk = (lane >> 4) << 1;

    const int row0 = blockIdx.y * 128 + wm * 64;
    const int col0 = blockIdx.x * 256 + wn * 64;

    const uint32_t voff = (uint32_t)(lm * EC + lk);
    const float* abase = ec    + (size_t)row0 * EC;
    const float* bbase = delta + (size_t)col0 * EC;

    v8f acc[4][4];
#pragma unroll
    for (int i = 0; i < 4; ++i)
#pragma unroll
        for (int j = 0; j < 4; ++j) acc[i][j] = (v8f){};

    for (int kb = 0; kb < EC; kb += 32) {
#pragma unroll
        for (int kk = 0; kk < 32; kk += 4)
            gemm_step_rowmajor(abase + kb + kk, bbase + kb + kk,
                               voff, voff, acc);
    }

    float lsum = 0.f;
#pragma unroll
    for (int mi = 0; mi < 4; ++mi)
#pragma unroll
        for (int ni = 0; ni < 4; ++ni)
#pragma unroll
            for (int r = 0; r < 8; ++r) {
                const float v = acc[mi][ni][r];
                lsum += v * v;
            }

    __shared__ float red[256];
    red[threadIdx.x] = lsum;
    __syncthreads();
#pragma unroll
    for (int s = 128; s > 0; s >>= 1) {
        if (threadIdx.x < (unsigned)s) red[threadIdx.x] += red[threadIdx.x + s];
        __syncthreads();
    }
    if (threadIdx.x == 0)
        partials[blockIdx.y * gridDim.x + blockIdx.x] = red[0];
}

// ---------------------------------------------------------------------------
// Kernel 5: fixed-order final sum of partials -> losses (deterministic)
// ---------------------------------------------------------------------------
__global__ void k_finalize(const float* __restrict__ partials, int n,
                           float* __restrict__ out_losses)
{
    if (threadIdx.x == 0 && blockIdx.x == 0) {
        float s = 0.f;
        for (int i = 0; i < n; ++i) s += partials[i];
        const float loss = s / (float)((size_t)BATCH * CA3);
        out_losses[0] = 0.0f;   // dg_ca3_loss (identically zero)
        out_losses[1] = loss;   // ec_ca3_loss
        out_losses[2] = loss;   // pc_cue_loss (dg cancels -> identical)
    }
}

extern "C" void kernel_launch(void* const* d_in, const int* in_sizes, int n_in,
                              void* d_out, int out_size, void* d_ws, size_t ws_size,
                              hipStream_t stream)
{
    (void)in_sizes; (void)n_in; (void)out_size; (void)ws_size;
    const float* ec = (const float*)d_in[0];   // (BATCH, EC)
    const float* dg = (const float*)d_in[1];   // (BATCH, CA3)
    const float* W  = (const float*)d_in[2];   // (CA3, EC)
    float* out = (float*)d_out;                // pc_cue (BATCH*CA3) + 3 losses

    float* delta    = (float*)d_ws;                    // (CA3, EC)  32 MB
    float* colmean  = delta + (size_t)CA3 * EC;        // CA3 floats
    float* partials = colmean + CA3;                   // 64 floats

    // 1) pre_pc_cue = ec @ W^T + dg   -> d_out
    k_gemm_pre<<<dim3(CA3 / 256, BATCH / 128), 256, 0, stream>>>(ec, dg, W, out);
    // 2) column mean of pc
    k_colmean<<<CA3 / 256, 256, 0, stream>>>(out, colmean);
    // 3) delta = LR * d_ec
    k_delta<<<dim3(EC / 256, CA3 / 128), 256, 0, stream>>>(out, ec, W, colmean, delta);
    // 4) diff = ec @ delta^T ; per-block sum of squares
    k_loss<<<dim3(CA3 / 256, BATCH / 128), 256, 0, stream>>>(ec, delta, partials);
    // 5) losses
    k_finalize<<<1, 64, 0, stream>>>(partials, 64, out + (size_t)BATCH * CA3);
}